// Foveater_16277926052663
// MI455X (gfx1250) — compile-verified
//
#include <hip/hip_runtime.h>
#include <math.h>

// ---------------------------------------------------------------------------
// Foveater: multi-scale patch resize + embedding GEMM for MI455X (gfx1250).
//   Kernel 1: per-batch 64x64 window -> three 16x16 resized patches (fp32)
//   Kernel 2: pack W_emb rows pairwise for b64 B-fetches
//   Kernel 3: [1024,2304] @ [2304,768] via V_WMMA_F32_16X16X4_F32 (exact fp32)
// ---------------------------------------------------------------------------

typedef float v2f __attribute__((ext_vector_type(2)));
typedef float v8f __attribute__((ext_vector_type(8)));

#define IMG_AXIS 128
#define EMB_DIM  768
#define IN_FEAT  2304   // 16*16*3ps*3ch
#define BATCH    1024

// ===========================================================================
// Kernel 1: patch extraction + antialiased bilinear resize (align_corners,
// triangle filter, float64 weight computation to match numpy exactly).
// The 64x64 window centered at the pixel contains the 32x32 (offset 16) and
// 16x16 (offset 24) windows, so we load global memory once per channel.
// ===========================================================================
__global__ __launch_bounds__(256) void foveate_kernel(
    const float* __restrict__ img,
    const int*   __restrict__ pixel_x,
    const int*   __restrict__ pixel_y,
    float*       __restrict__ tok)            // [BATCH, IN_FEAT]
{
    __shared__ float W32[16][32];
    __shared__ float W64[16][64];
    __shared__ float patch[64][64];           // ps=64 window (zero padded)
    __shared__ float tmp32[16][32];
    __shared__ float tmp64[16][64];

    const int b   = blockIdx.x;
    const int tid = threadIdx.x;
    const int px  = pixel_x[b];
    const int py  = pixel_y[b];

    // ---- resize matrices in double precision (matches np.float64 -> f32) ----
    if (tid < 16) {
        const double scale  = 31.0 / 15.0;    // (32-1)/(16-1), support = scale
        const double center = (double)tid * scale;
        double row[32]; double s = 0.0;
        for (int j = 0; j < 32; ++j) {
            double w = 1.0 - fabs((double)j - center) / scale;
            w = (w > 0.0) ? w : 0.0;
            row[j] = w; s += w;
        }
        for (int j = 0; j < 32; ++j) W32[tid][j] = (float)(row[j] / s);
    } else if (tid < 32) {
        const int o = tid - 16;
        const double scale  = 63.0 / 15.0;    // (64-1)/(16-1)
        const double center = (double)o * scale;
        double row[64]; double s = 0.0;
        for (int j = 0; j < 64; ++j) {
            double w = 1.0 - fabs((double)j - center) / scale;
            w = (w > 0.0) ? w : 0.0;
            row[j] = w; s += w;
        }
        for (int j = 0; j < 64; ++j) W64[o][j] = (float)(row[j] / s);
    }

    float* row_out = tok + (size_t)b * IN_FEAT;

    for (int ch = 0; ch < 3; ++ch) {
        __syncthreads();   // W matrices ready / previous channel consumed

        // ---- load 64x64 window, top-left at (py-32, px-32), zero padded ----
        const float* imc = img + ((size_t)(b * 3 + ch)) * (IMG_AXIS * IMG_AXIS);
        #pragma unroll
        for (int i = 0; i < 16; ++i) {
            const int e  = tid + 256 * i;
            const int r  = e >> 6;
            const int c  = e & 63;
            const int rr = py - 32 + r;
            const int cc = px - 32 + c;
            float v = 0.0f;
            if (rr >= 0 && rr < IMG_AXIS && cc >= 0 && cc < IMG_AXIS)
                v = imc[rr * IMG_AXIS + cc];
            patch[r][c] = v;
        }
        __syncthreads();

        // ---- ps=16: identity (center 16x16 at offset 24) ----
        {
            const int o = tid >> 4, p = tid & 15;
            row_out[(0 * 3 + ch) * 256 + o * 16 + p] = patch[24 + o][24 + p];
        }
        // ---- stage 1: tmp[o][j] = sum_i Wm[o][i] * patch[i][j] ----
        for (int e = tid; e < 16 * 32; e += 256) {   // ps=32 (window offset 16)
            const int o = e >> 5, j = e & 31;
            float s = 0.0f;
            for (int i = 0; i < 32; ++i) s = fmaf(W32[o][i], patch[16 + i][16 + j], s);
            tmp32[o][j] = s;
        }
        for (int e = tid; e < 16 * 64; e += 256) {   // ps=64
            const int o = e >> 6, j = e & 63;
            float s = 0.0f;
            for (int i = 0; i < 64; ++i) s = fmaf(W64[o][i], patch[i][j], s);
            tmp64[o][j] = s;
        }
        __syncthreads();

        // ---- stage 2: out[o][p] = sum_j tmp[o][j] * Wm[p][j] ----
        {
            const int o = tid >> 4, p = tid & 15;
            float s32 = 0.0f;
            for (int j = 0; j < 32; ++j) s32 = fmaf(tmp32[o][j], W32[p][j], s32);
            row_out[(1 * 3 + ch) * 256 + o * 16 + p] = s32;
            float s64 = 0.0f;
            for (int j = 0; j < 64; ++j) s64 = fmaf(tmp64[o][j], W64[p][j], s64);
            row_out[(2 * 3 + ch) * 256 + o * 16 + p] = s64;
        }
    }
}

// ===========================================================================
// Kernel 2: pack W_emb [2304,768] -> Wpack [1152][768][2]:
//   Wpack[kp][n] = { W[2kp][n], W[2kp+1][n] }
// so the WMMA B operand (two K-rows, same column) is one global_load_b64.
// ===========================================================================
__global__ __launch_bounds__(256) void pack_w_kernel(
    const float* __restrict__ W, float* __restrict__ Wp)
{
    const int idx = blockIdx.x * 256 + threadIdx.x;     // over (IN_FEAT/2)*EMB_DIM
    const int kp  = idx / EMB_DIM;
    const int n   = idx - kp * EMB_DIM;
    v2f v;
    v.x = W[(size_t)(2 * kp)     * EMB_DIM + n];
    v.y = W[(size_t)(2 * kp + 1) * EMB_DIM + n];
    ((v2f*)Wp)[idx] = v;
}

// ===========================================================================
// Kernel 3: out = tok @ W_emb + b_emb + pos_emb[py,px]  (exact fp32 via WMMA)
// Block = 128 threads (4 waves). Each wave owns a 16(M) x 64(N) slab:
//   4 accumulators; A pair loaded once per K-step (b64), reused 4x;
//   B fetched as b64 from the packed layout (PACKED=true).
// Grid = (64 m-tiles, 3); wave w handles N group (blockIdx.y*4 + w)*64.
// WMMA f32 16x16x4 layouts:
//   A: lane L -> a.x=A[L%16][k+2*(L/16)], a.y=A[L%16][k+2*(L/16)+1]
//   B: lane L -> b.x=B[k+2*(L/16)][L%16], b.y=B[k+2*(L/16)+1][L%16]
//   D: vgpr r -> D[r + 8*(L/16)][L%16]
// ===========================================================================
template <bool PACKED>
__global__ __launch_bounds__(128) void embed_gemm_kernel(
    const float* __restrict__ tok,      // [BATCH, IN_FEAT]
    const float* __restrict__ W,        // PACKED ? [IN_FEAT/2,EMB_DIM,2] : [IN_FEAT,EMB_DIM]
    const float* __restrict__ b_emb,    // [EMB_DIM]
    const float* __restrict__ pos_emb,  // [IMG_AXIS, IMG_AXIS, EMB_DIM]
    const int*   __restrict__ pixel_x,
    const int*   __restrict__ pixel_y,
    float*       __restrict__ out)      // [BATCH, EMB_DIM]
{
    const int lane = threadIdx.x & 31;
    const int wave = threadIdx.x >> 5;
    const int m0   = blockIdx.x * 16;
    const int n0   = (blockIdx.y * 4 + wave) * 64;
    const int half = lane >> 4;          // 0: lanes 0-15, 1: lanes 16-31
    const int mr   = lane & 15;

    v8f acc[4];
    #pragma unroll
    for (int t = 0; t < 4; ++t) acc[t] = (v8f){0.f,0.f,0.f,0.f,0.f,0.f,0.f,0.f};

    const float* arow = tok + (size_t)(m0 + mr) * IN_FEAT + 2 * half;

    for (int k = 0; k < IN_FEAT; k += 4) {
        const v2f a = *(const v2f*)(arow + k);           // A[m][ka], A[m][ka+1]
        if (PACKED) {
            // pair row kp = ka/2 = k/2 + half holds W rows (ka, ka+1)
            const v2f* bp = (const v2f*)W
                          + ((size_t)(k / 2 + half) * EMB_DIM + n0 + mr);
            #pragma unroll
            for (int t = 0; t < 4; ++t) {
                const v2f bb = bp[16 * t];               // one b64 per tile
                acc[t] = __builtin_amdgcn_wmma_f32_16x16x4_f32(
                    false, a, false, bb, (short)0, acc[t], false, false);
            }
        } else {
            const float* bp0 = W + (size_t)(k + 2 * half) * EMB_DIM + n0 + mr;
            const float* bp1 = bp0 + EMB_DIM;
            #pragma unroll
            for (int t = 0; t < 4; ++t) {
                v2f bb;
                bb.x = bp0[16 * t];
                bb.y = bp1[16 * t];
                acc[t] = __builtin_amdgcn_wmma_f32_16x16x4_f32(
                    false, a, false, bb, (short)0, acc[t], false, false);
            }
        }
    }

    // epilogue: + bias + pos_emb gather
    #pragma unroll
    for (int t = 0; t < 4; ++t) {
        const int col  = n0 + 16 * t + mr;
        const float bias = b_emb[col];
        #pragma unroll
        for (int r = 0; r < 8; ++r) {
            const int mrow = m0 + r + 8 * half;
            const size_t pidx =
                ((size_t)pixel_y[mrow] * IMG_AXIS + pixel_x[mrow]) * EMB_DIM + col;
            out[(size_t)mrow * EMB_DIM + col] = acc[t][r] + bias + pos_emb[pidx];
        }
    }
}

// ===========================================================================
extern "C" void kernel_launch(void* const* d_in, const int* in_sizes, int n_in,
                              void* d_out, int out_size, void* d_ws, size_t ws_size,
                              hipStream_t stream) {
    const float* img     = (const float*)d_in[0];   // [1024,3,128,128]
    const int*   pixel_x = (const int*)  d_in[1];   // [1024]
    const int*   pixel_y = (const int*)  d_in[2];   // [1024]
    const float* W_emb   = (const float*)d_in[3];   // [2304,768]
    const float* b_emb   = (const float*)d_in[4];   // [768]
    const float* pos_emb = (const float*)d_in[5];   // [128,128,768]
    float*       out     = (float*)d_out;           // [1024,768]

    const size_t TOK_BYTES  = (size_t)BATCH * IN_FEAT * sizeof(float);     // 9.4 MB
    const size_t PACK_BYTES = (size_t)IN_FEAT * EMB_DIM * sizeof(float);   // 7.1 MB

    float* tok = (float*)d_ws;

    foveate_kernel<<<BATCH, 256, 0, stream>>>(img, pixel_x, pixel_y, tok);

    dim3 grid(BATCH / 16, EMB_DIM / (4 * 64));      // (64, 3), 4 waves/block

    if (ws_size >= TOK_BYTES + PACK_BYTES) {
        float* Wpack = (float*)((char*)d_ws + TOK_BYTES);
        pack_w_kernel<<<(IN_FEAT / 2) * EMB_DIM / 256, 256, 0, stream>>>(W_emb, Wpack);
        embed_gemm_kernel<true><<<grid, 128, 0, stream>>>(
            tok, Wpack, b_emb, pos_emb, pixel_x, pixel_y, out);
    } else {
        embed_gemm_kernel<false><<<grid, 128, 0, stream>>>(
            tok, W_emb, b_emb, pos_emb, pixel_x, pixel_y, out);
    }
}